// BiggerBirdBartForSequenceClassification_8830452760954
// MI455X (gfx1250) — compile-verified
//
#include <hip/hip_runtime.h>
#include <stdint.h>

typedef float v2f __attribute__((ext_vector_type(2)));
typedef float v8f __attribute__((ext_vector_type(8)));

#define T_    2048
#define D_    64
#define H_    16
#define B_    1
#define BH_   16
#define FW_   64
#define G_    6
#define TEL_  3
#define P_    24
#define TOPU_ 16
#define NEXT_ (G_ + TEL_)   // 9 irregular keys per query
#define SCALE_ 0.125f       // 1/sqrt(64)
#define ALPHA_ 0.15f
#define TAU_   8.0f

// ---------------------------------------------------------------------------
// Kernel 1: global chooser. One block per (b,h). Computes the 6 global key
// positions per head via prototype scoring + top-16 + greedy facility location.
// ---------------------------------------------------------------------------
__global__ __launch_bounds__(256) void bb_chooser(const float* __restrict__ q,
                                                  const float* __restrict__ k,
                                                  int* __restrict__ gidx) {
  const int bh = blockIdx.x;
  const float* Qb = q + (size_t)bh * T_ * D_;
  const float* Kb = k + (size_t)bh * T_ * D_;

  __shared__ float Qp[P_][68];     // normalized prototypes
  __shared__ float util[T_];
  __shared__ float redV[256];
  __shared__ int   redI[256];
  __shared__ float Ssub[TOPU_][P_];
  __shared__ int   topIdx[TOPU_];

  const int tid = threadIdx.x;

  // --- normalized query prototypes at round(linspace(0, T-1, 24)) ---
  if (tid < P_) {
    int pos = (int)floorf((float)(T_ - 1) * (float)tid / (float)(P_ - 1) + 0.5f);
    float ss = 0.f;
    for (int d = 0; d < D_; ++d) { float x = Qb[pos * D_ + d]; ss += x * x; }
    float inv = 1.0f / fmaxf(sqrtf(ss), 1e-6f);
    for (int d = 0; d < D_; ++d) Qp[tid][d] = Qb[pos * D_ + d] * inv;
  }
  __syncthreads();

  // --- per-key utility ---
  for (int t = tid; t < T_; t += 256) {
    float ss = 0.f;
    for (int d = 0; d < D_; ++d) { float x = Kb[t * D_ + d]; ss += x * x; }
    float inv = 1.0f / fmaxf(sqrtf(ss), 1e-6f);
    float s[P_];
    for (int p = 0; p < P_; ++p) s[p] = 0.f;
    for (int d = 0; d < D_; ++d) {
      float kv = Kb[t * D_ + d] * inv;
      for (int p = 0; p < P_; ++p) s[p] += kv * Qp[p][d];
    }
    float sum = 0.f, mx = 0.f, sumsq = 0.f;
    for (int p = 0; p < P_; ++p) {
      s[p] = fmaxf(s[p], 0.f);
      sum += s[p]; sumsq += s[p] * s[p]; mx = fmaxf(mx, s[p]);
    }
    float mean = sum / (float)P_;
    float tmp[P_];
    for (int p = 0; p < P_; ++p) tmp[p] = s[p];
    float tsum = 0.f;                      // top-5 mean (kq = round(24*0.2) = 5)
    for (int r = 0; r < 5; ++r) {
      float bv = -1.f; int bi = 0;
      for (int p = 0; p < P_; ++p) if (tmp[p] > bv) { bv = tmp[p]; bi = p; }
      tsum += bv; tmp[bi] = -1.f;
    }
    float var = (sumsq - (float)P_ * mean * mean) / (float)(P_ - 1);
    var = fmaxf(var, 0.f);
    util[t] = 1.0f * mean + 0.6f * mx + 0.4f * (tsum * 0.2f) + 0.2f * sqrtf(var);
  }
  __syncthreads();

  // --- top-16 by iterative parallel argmax (lower index wins ties) ---
  for (int it = 0; it < TOPU_; ++it) {
    float bv = -1e30f; int bi = 0;
    for (int t = tid; t < T_; t += 256) {
      float val = util[t];
      if (val > bv) { bv = val; bi = t; }
    }
    redV[tid] = bv; redI[tid] = bi;
    __syncthreads();
    for (int ofs = 128; ofs > 0; ofs >>= 1) {
      if (tid < ofs) {
        float ov = redV[tid + ofs]; int oi = redI[tid + ofs];
        if (ov > redV[tid] || (ov == redV[tid] && oi < redI[tid])) {
          redV[tid] = ov; redI[tid] = oi;
        }
      }
      __syncthreads();
    }
    if (tid == 0) { topIdx[it] = redI[0]; util[redI[0]] = -1e30f; }
    __syncthreads();
  }

  // --- recompute S rows for the 16 candidates ---
  for (int i = tid; i < TOPU_ * P_; i += 256) {
    int c = i / P_, p = i % P_;
    int t = topIdx[c];
    float ss = 0.f;
    for (int d = 0; d < D_; ++d) { float x = Kb[t * D_ + d]; ss += x * x; }
    float inv = 1.0f / fmaxf(sqrtf(ss), 1e-6f);
    float dot = 0.f;
    for (int d = 0; d < D_; ++d) dot += Kb[t * D_ + d] * inv * Qp[p][d];
    Ssub[c][p] = fmaxf(dot, 0.f);
  }
  __syncthreads();

  // --- greedy facility location, G=6 picks (serial, tiny) ---
  if (tid == 0) {
    float m[P_]; bool blocked[TOPU_];
    for (int p = 0; p < P_; ++p) m[p] = 0.f;
    for (int j = 0; j < TOPU_; ++j) blocked[j] = false;
    for (int g = 0; g < G_; ++g) {
      float best = -2e9f; int bj = 0;
      for (int j = 0; j < TOPU_; ++j) {
        float gain = -1e9f;
        if (!blocked[j]) {
          gain = 0.f;
          for (int p = 0; p < P_; ++p) gain += fmaxf(Ssub[j][p] - m[p], 0.f);
        }
        if (gain > best) { best = gain; bj = j; }
      }
      blocked[bj] = true;
      for (int p = 0; p < P_; ++p) m[p] = fmaxf(m[p], Ssub[bj][p]);
      gidx[bh * G_ + g] = topIdx[bj];
    }
  }
}

// ---------------------------------------------------------------------------
// Kernel 2: gathered attention. One wave32 per 16-query tile.
//  - V union staged via async global->LDS (ASYNCcnt), overlapped with QK^T,
//    extras scoring and softmax; waited just before the PV matmul.
//  - WMMA f32 16x16x4 for both matmuls; scalar path for globals/teleports.
// ---------------------------------------------------------------------------
__global__ __launch_bounds__(32) void bb_attn(const float* __restrict__ q,
                                              const float* __restrict__ k,
                                              const float* __restrict__ v,
                                              const int* __restrict__ tp,
                                              const int* __restrict__ gidx,
                                              float* __restrict__ out) {
  const int bh   = blockIdx.y;
  const int t0   = blockIdx.x * 16;
  const int lane = threadIdx.x;
  const int m    = lane & 15;
  const int hi   = lane >> 4;

  const float* Qb = q + (size_t)bh * T_ * D_;
  const float* Kb = k + (size_t)bh * T_ * D_;
  const float* Vb = v + (size_t)bh * T_ * D_;

  // Row pads chosen for conflict-free column-wise fragment reads + 16B rows.
  __shared__ float Qs[16][68];
  __shared__ float Ku[80][68];
  __shared__ float Vu[80][80];
  __shared__ float Sm[16][84];   // scores -> probs (in place)
  __shared__ float Om[16][68];
  __shared__ float esc[16][12];  // extras scores -> probs
  __shared__ int   eix[16][12];
  __shared__ float ptab[80];     // alpha * exp(-d/tau) lookup, window distances
  __shared__ float pred[16][4];  // partial max [0..1], partial sum [2..3]
  __shared__ float gmxs[16], ginv[16];

  int smin = t0 - FW_ / 2;
  if (smin < 0) smin = 0;
  if (smin > T_ - FW_) smin = T_ - FW_;

  // ---- issue async V-union staging first: overlaps everything up to PV ----
  for (int i = lane; i < 80 * 16; i += 32) {
    int row = i >> 4, f = i & 15;
    int key = smin + row; if (key > T_ - 1) key = T_ - 1;
    unsigned long long ga = (unsigned long long)(uintptr_t)(Vb + key * D_ + f * 4);
    unsigned int la = (unsigned int)(uintptr_t)&Vu[row][f * 4];  // LDS offset = addr[31:0]
    asm volatile("global_load_async_to_lds_b128 %0, %1, off"
                 :: "v"(la), "v"(ga) : "memory");
  }

  // positional-prior lookup table (window distances are <= 79)
  for (int i = lane; i < 80; i += 32)
    ptab[i] = ALPHA_ * __expf(-(float)i / TAU_);

  // irregular key indices: 6 globals (shared) + 3 teleports (per query)
  for (int i = lane; i < 16 * NEXT_; i += 32) {
    int qq = i / NEXT_, e = i % NEXT_;
    int t = t0 + qq;
    int key = (e < G_) ? gidx[bh * G_ + e]
                       : tp[((size_t)bh * T_ + t) * TEL_ + (e - G_)];
    eix[qq][e] = key;
    __builtin_prefetch(Kb + key * D_, 0, 3);   // scattered rows -> prefetch
    __builtin_prefetch(Vb + key * D_, 0, 3);
  }

  // stage Q tile (16x64) and K union (80 rows) synchronously
  for (int i = lane; i < 16 * 16; i += 32) {
    int row = i >> 4, f = i & 15;
    *(float4*)(&Qs[row][f * 4]) = *(const float4*)(Qb + (t0 + row) * D_ + f * 4);
  }
  for (int i = lane; i < 80 * 16; i += 32) {
    int row = i >> 4, f = i & 15;
    int key = smin + row; if (key > T_ - 1) key = T_ - 1;
    *(float4*)(&Ku[row][f * 4]) = *(const float4*)(Kb + key * D_ + f * 4);
  }
  __syncthreads();

  // per-query window geometry
  int sq = t0 + m - FW_ / 2;
  if (sq < 0) sq = 0;
  if (sq > T_ - FW_) sq = T_ - FW_;
  const int off = sq - smin;        // window start within the 80-wide union

  // irregular-key scores (scaled + positional prior)
  for (int i = lane; i < 16 * NEXT_; i += 32) {
    int qq = i / NEXT_, e = i % NEXT_;
    int key = eix[qq][e];
    float dot = 0.f;
    for (int d = 0; d < D_; ++d) dot += Qs[qq][d] * Kb[key * D_ + d];
    float dist = fabsf((float)(key - (t0 + qq)));
    esc[qq][e] = dot * SCALE_ + ALPHA_ * __expf(-dist / TAU_);
  }

  // ---- S(16x80) = Q(16x64) @ K_union^T via V_WMMA_F32_16X16X4_F32 ----
  v2f aQ[16];
#pragma unroll
  for (int s = 0; s < 16; ++s) {
    aQ[s].x = Qs[m][4 * s + 2 * hi];
    aQ[s].y = Qs[m][4 * s + 2 * hi + 1];
  }
#pragma unroll
  for (int nt = 0; nt < 5; ++nt) {
    const int n0 = nt * 16;
    v8f acc = {0.f, 0.f, 0.f, 0.f, 0.f, 0.f, 0.f, 0.f};
#pragma unroll
    for (int s = 0; s < 16; ++s) {
      v2f bK;
      bK.x = Ku[n0 + m][4 * s + 2 * hi];
      bK.y = Ku[n0 + m][4 * s + 2 * hi + 1];
      acc = __builtin_amdgcn_wmma_f32_16x16x4_f32(false, aQ[s], false, bK,
                                                  (short)0, acc, false, false);
    }
#pragma unroll
    for (int r = 0; r < 8; ++r) Sm[r + 8 * hi][n0 + m] = acc[r];
  }
  __syncthreads();

  // ---- softmax over 64 window + 9 irregular entries, all 32 lanes ----
  // pass 1: scale + prior (table lookup), partial max over half-window
  {
    const int t = t0 + m;
    float mx = -1e30f;
    for (int jj = 0; jj < 32; ++jj) {
      int j = 32 * hi + jj;
      int u = off + j;
      int d = sq + j - t; d = (d < 0) ? -d : d;     // <= 79
      float sc = Sm[m][u] * SCALE_ + ptab[d];
      Sm[m][u] = sc;
      mx = fmaxf(mx, sc);
    }
    pred[m][hi] = mx;
  }
  __syncthreads();
  if (lane < 16) {
    float mx = fmaxf(pred[lane][0], pred[lane][1]);
    for (int e = 0; e < NEXT_; ++e) mx = fmaxf(mx, esc[lane][e]);
    gmxs[lane] = mx;
  }
  __syncthreads();
  // pass 2: partial exp-sums
  {
    const float mx = gmxs[m];
    float s = 0.f;
    for (int jj = 0; jj < 32; ++jj)
      s += __expf(Sm[m][off + 32 * hi + jj] - mx);
    pred[m][2 + hi] = s;
  }
  __syncthreads();
  if (lane < 16) {
    const float mx = gmxs[lane];
    float sum = pred[lane][2] + pred[lane][3];
    for (int e = 0; e < NEXT_; ++e) sum += __expf(esc[lane][e] - mx);
    const float inv = 1.0f / sum;
    ginv[lane] = inv;
    for (int e = 0; e < NEXT_; ++e) esc[lane][e] = __expf(esc[lane][e] - mx) * inv;
  }
  __syncthreads();
  // pass 3: write padded prob row (zeros outside each query's 64-slice)
  {
    const float mx = gmxs[m], inv = ginv[m];
    for (int uu = 0; uu < 40; ++uu) {
      int u = 40 * hi + uu;
      float pv = 0.f;
      if (u >= off && u < off + FW_) pv = __expf(Sm[m][u] - mx) * inv;
      Sm[m][u] = pv;
    }
  }

  // V union must have landed in LDS before the PV matmul
  asm volatile("s_wait_asynccnt 0" ::: "memory");
  __syncthreads();

  // ---- O(16x64) = P(16x80) @ V_union(80x64) via WMMA ----
  v2f aP[20];
#pragma unroll
  for (int s = 0; s < 20; ++s) {
    aP[s].x = Sm[m][4 * s + 2 * hi];
    aP[s].y = Sm[m][4 * s + 2 * hi + 1];
  }
#pragma unroll
  for (int dt = 0; dt < 4; ++dt) {
    const int n0 = dt * 16;
    v8f acc = {0.f, 0.f, 0.f, 0.f, 0.f, 0.f, 0.f, 0.f};
#pragma unroll
    for (int s = 0; s < 20; ++s) {
      v2f bV;
      bV.x = Vu[4 * s + 2 * hi][n0 + m];
      bV.y = Vu[4 * s + 2 * hi + 1][n0 + m];
      acc = __builtin_amdgcn_wmma_f32_16x16x4_f32(false, aP[s], false, bV,
                                                  (short)0, acc, false, false);
    }
#pragma unroll
    for (int r = 0; r < 8; ++r) Om[r + 8 * hi][n0 + m] = acc[r];
  }
  __syncthreads();

  // ---- add irregular-key V contributions, write [B, T, H*D] ----
  const int b = bh / H_, h = bh % H_;
  for (int i = lane; i < 16 * D_; i += 32) {
    int qq = i >> 6, d = i & 63;
    float val = Om[qq][d];
    for (int e = 0; e < NEXT_; ++e)
      val += esc[qq][e] * Vb[eix[qq][e] * D_ + d];
    out[(((size_t)b * T_ + (t0 + qq)) * H_ + h) * D_ + d] = val;
  }
}

// ---------------------------------------------------------------------------
extern "C" void kernel_launch(void* const* d_in, const int* in_sizes, int n_in,
                              void* d_out, int out_size, void* d_ws, size_t ws_size,
                              hipStream_t stream) {
  (void)in_sizes; (void)n_in; (void)out_size; (void)ws_size;
  const float* q  = (const float*)d_in[0];
  const float* k  = (const float*)d_in[1];
  const float* v  = (const float*)d_in[2];
  const int*   tp = (const int*)d_in[3];
  int* gidx = (int*)d_ws;                 // BH*G ints of scratch
  float* out = (float*)d_out;

  bb_chooser<<<BH_, 256, 0, stream>>>(q, k, gidx);
  bb_attn<<<dim3(T_ / 16, BH_), 32, 0, stream>>>(q, k, v, tp, gidx, out);
}